// ACOPFGNN_39694087749651
// MI455X (gfx1250) — compile-verified
//
#include <hip/hip_runtime.h>
#include <hip/hip_bf16.h>
#include <math.h>

typedef __attribute__((ext_vector_type(2))) float v2f;
typedef __attribute__((ext_vector_type(8))) float v8f;

#define EDG 150000
#define D_STRIDE 192

__device__ __forceinline__ float eluf(float x) { return x > 0.f ? x : (__expf(x) - 1.f); }
__device__ __forceinline__ unsigned f2o(float f) {
  unsigned u = __float_as_uint(f);
  return (u & 0x80000000u) ? ~u : (u | 0x80000000u);
}
__device__ __forceinline__ float o2f(unsigned e) {
  return (e & 0x80000000u) ? __uint_as_float(e & 0x7FFFFFFFu) : __uint_as_float(~e);
}

// ---------------------------------------------------------------------------
// K0: per (layer, edge-type) derived weight reductions (~150 scalars each).
// Slots: 0..63 M8[p*8+r]=Wq[p].Wk[r]; 64..71 u=Wq.bk; 72..79 w=Wk.bq; 80 c0=bq.bk
// 81..88 Wq.We0; 89..96 Wq.We1; 97..104 Wq.be; 105 bq.We0; 106 bq.We1; 107 bq.be
// 108..115 sum_{h<64}Wv; 116..123 sum_{h>=64}Wv; 124/125 bv half sums
// 126..129 Eh[r*2+half]=sum_half We[r]; 130/131 be half sums
// 132..139 / 140..147 Ws half col-sums; 148/149 bs half sums
// ---------------------------------------------------------------------------
__global__ void k_derive(const float* __restrict__ Wq, const float* __restrict__ bq,
                         const float* __restrict__ Wk, const float* __restrict__ bk,
                         const float* __restrict__ Wv, const float* __restrict__ bv,
                         const float* __restrict__ We, const float* __restrict__ be,
                         const float* __restrict__ Ws, const float* __restrict__ bs,
                         float* __restrict__ der, unsigned* __restrict__ mm) {
  int it = blockIdx.x;
  int tid = threadIdx.x;
  if (it == 0) {
    if (tid == 150) mm[0] = 0xFFFFFFFFu;  // global min (encoded)
    if (tid == 151) mm[1] = 0u;           // global max (encoded)
  }
  const float* wq  = Wq + it * 1024;
  const float* wk  = Wk + it * 1024;
  const float* wv  = Wv + it * 1024;
  const float* wsk = Ws + it * 1024;
  const float* we  = We + it * 256;
  const float* vbq = bq + it * 128;
  const float* vbk = bk + it * 128;
  const float* vbv = bv + it * 128;
  const float* vbe = be + it * 128;
  const float* vbs = bs + it * 128;
  float r = 0.f;
  int slot = -1;
  if (tid < 64) {
    int a = tid >> 3, b = tid & 7;
    for (int h = 0; h < 128; ++h) r += wq[a * 128 + h] * wk[b * 128 + h];
    slot = tid;
  } else if (tid < 72) {
    int a = tid - 64;
    for (int h = 0; h < 128; ++h) r += wq[a * 128 + h] * vbk[h];
    slot = 64 + a;
  } else if (tid < 80) {
    int a = tid - 72;
    for (int h = 0; h < 128; ++h) r += wk[a * 128 + h] * vbq[h];
    slot = 72 + a;
  } else if (tid == 80) {
    for (int h = 0; h < 128; ++h) r += vbq[h] * vbk[h];
    slot = 80;
  } else if (tid < 89) {
    int a = tid - 81;
    for (int h = 0; h < 128; ++h) r += wq[a * 128 + h] * we[h];
    slot = 81 + a;
  } else if (tid < 97) {
    int a = tid - 89;
    for (int h = 0; h < 128; ++h) r += wq[a * 128 + h] * we[128 + h];
    slot = 89 + a;
  } else if (tid < 105) {
    int a = tid - 97;
    for (int h = 0; h < 128; ++h) r += wq[a * 128 + h] * vbe[h];
    slot = 97 + a;
  } else if (tid == 105) {
    for (int h = 0; h < 128; ++h) r += vbq[h] * we[h];
    slot = 105;
  } else if (tid == 106) {
    for (int h = 0; h < 128; ++h) r += vbq[h] * we[128 + h];
    slot = 106;
  } else if (tid == 107) {
    for (int h = 0; h < 128; ++h) r += vbq[h] * vbe[h];
    slot = 107;
  } else if (tid < 116) {
    int a = tid - 108;
    for (int h = 0; h < 64; ++h) r += wv[a * 128 + h];
    slot = 108 + a;
  } else if (tid < 124) {
    int a = tid - 116;
    for (int h = 64; h < 128; ++h) r += wv[a * 128 + h];
    slot = 116 + a;
  } else if (tid == 124) {
    for (int h = 0; h < 64; ++h) r += vbv[h];
    slot = 124;
  } else if (tid == 125) {
    for (int h = 64; h < 128; ++h) r += vbv[h];
    slot = 125;
  } else if (tid < 130) {
    int idx = tid - 126, rr = idx >> 1, hf = idx & 1;
    for (int h = 0; h < 64; ++h) r += we[rr * 128 + hf * 64 + h];
    slot = 126 + idx;
  } else if (tid == 130) {
    for (int h = 0; h < 64; ++h) r += vbe[h];
    slot = 130;
  } else if (tid == 131) {
    for (int h = 64; h < 128; ++h) r += vbe[h];
    slot = 131;
  } else if (tid < 140) {
    int a = tid - 132;
    for (int h = 0; h < 64; ++h) r += wsk[a * 128 + h];
    slot = 132 + a;
  } else if (tid < 148) {
    int a = tid - 140;
    for (int h = 64; h < 128; ++h) r += wsk[a * 128 + h];
    slot = 140 + a;
  } else if (tid == 148) {
    for (int h = 0; h < 64; ++h) r += vbs[h];
    slot = 148;
  } else if (tid == 149) {
    for (int h = 64; h < 128; ++h) r += vbs[h];
    slot = 149;
  }
  if (slot >= 0) der[it * D_STRIDE + slot] = r;
}

// ---------------------------------------------------------------------------
// K1: per-node precompute (dest + src sides) + segment buffer init + skip->g.
// ---------------------------------------------------------------------------
__global__ void k_prep(const float* __restrict__ xd, const float* __restrict__ cd,
                       const float* __restrict__ xs, const float* __restrict__ cs,
                       const float* __restrict__ dv, float* __restrict__ dstb,
                       float* __restrict__ srcb, unsigned* __restrict__ mseg,
                       float* __restrict__ den, float* __restrict__ g, int gBase,
                       int Nd, int Ns) {
  int n = blockIdx.x * blockDim.x + threadIdx.x;
  if (n < Nd) {
    float4 t0 = ((const float4*)xd)[n];
    float4 t1 = ((const float4*)cd)[n];
    float xv[8] = {t0.x, t0.y, t0.z, t0.w, t1.x, t1.y, t1.z, t1.w};
    float* o = dstb + n * 16;
#pragma unroll
    for (int a = 0; a < 8; ++a) {
      float s = 0.f;
#pragma unroll
      for (int b = 0; b < 8; ++b) s += xv[b] * dv[b * 8 + a];
      o[a] = s;  // A8 = xd^T M8
    }
    float dc = dv[80] + dv[107];
    float q0 = dv[105], q1 = dv[106];
    float s0 = dv[148], s1 = dv[149];
#pragma unroll
    for (int b = 0; b < 8; ++b) {
      dc += xv[b] * (dv[64 + b] + dv[97 + b]);
      q0 += xv[b] * dv[81 + b];
      q1 += xv[b] * dv[89 + b];
      s0 += xv[b] * dv[132 + b];
      s1 += xv[b] * dv[140 + b];
    }
    o[8] = dc;
    o[9] = q0;
    o[10] = q1;
    float* gp = g + (size_t)(gBase + n) * 2;
    gp[0] = s0;  // skip-connection half sums seed the group-sum buffer
    gp[1] = s1;
    mseg[n] = 0u;  // encoded -inf lower bound
    den[n] = 0.f;
  }
  if (n < Ns) {
    float4 t0 = ((const float4*)xs)[n];
    float4 t1 = ((const float4*)cs)[n];
    float xv[8] = {t0.x, t0.y, t0.z, t0.w, t1.x, t1.y, t1.z, t1.w};
    float sw = 0.f, v0 = dv[124], v1 = dv[125];
#pragma unroll
    for (int b = 0; b < 8; ++b) {
      sw += xv[b] * dv[72 + b];
      v0 += xv[b] * dv[108 + b];
      v1 += xv[b] * dv[116 + b];
    }
    float* o = srcb + n * 4;
    o[0] = sw;
    o[1] = v0;
    o[2] = v1;
  }
}

// K2: edge pass 1 — logits + segment max
__global__ void k_edge1(const int* __restrict__ src, const int* __restrict__ dst,
                        const float* __restrict__ ea, const float* __restrict__ xs,
                        const float* __restrict__ cs, const float* __restrict__ dstb,
                        const float* __restrict__ srcb, float* __restrict__ logit,
                        unsigned* __restrict__ mseg) {
  int e = blockIdx.x * blockDim.x + threadIdx.x;
  if (e >= EDG) return;
  int s = src[e], d = dst[e];
  float4 t0 = ((const float4*)xs)[s];
  float4 t1 = ((const float4*)cs)[s];
  const float* db = dstb + d * 16;
  float L = db[8] + srcb[s * 4];
  L += db[0] * t0.x + db[1] * t0.y + db[2] * t0.z + db[3] * t0.w +
       db[4] * t1.x + db[5] * t1.y + db[6] * t1.z + db[7] * t1.w;
  float e0 = ea[e * 2], e1 = ea[e * 2 + 1];
  L += e0 * db[9] + e1 * db[10];
  L *= 0.08838834764831845f;  // 1/sqrt(128)
  logit[e] = L;
  atomicMax(mseg + d, f2o(L));
}

// K3: edge pass 2 — exp + segment sum
__global__ void k_edge2(const int* __restrict__ dst, const float* __restrict__ logit,
                        const unsigned* __restrict__ mseg, float* __restrict__ pe,
                        float* __restrict__ den) {
  int e = blockIdx.x * blockDim.x + threadIdx.x;
  if (e >= EDG) return;
  int d = dst[e];
  float p = __expf(logit[e] - o2f(mseg[d]));
  pe[e] = p;
  atomicAdd(den + d, p);
}

// K4: edge pass 3 — alpha * (v-half-sums) scattered into group-sum buffer
__global__ void k_edge3(const int* __restrict__ src, const int* __restrict__ dst,
                        const float* __restrict__ ea, const float* __restrict__ srcb,
                        const float* __restrict__ den, const float* __restrict__ pe,
                        const float* __restrict__ dv, float* __restrict__ g, int gBase) {
  int e = blockIdx.x * blockDim.x + threadIdx.x;
  if (e >= EDG) return;
  int s = src[e], d = dst[e];
  float alpha = pe[e] / (den[d] + 1e-16f);
  float e0 = ea[e * 2], e1 = ea[e * 2 + 1];
  float eh0 = e0 * dv[126] + e1 * dv[128] + dv[130];
  float eh1 = e0 * dv[127] + e1 * dv[129] + dv[131];
  const float* sb = srcb + s * 4;
  float* gp = g + (size_t)(gBase + d) * 2;
  atomicAdd(gp + 0, alpha * (sb[1] + eh0));
  atomicAdd(gp + 1, alpha * (sb[2] + eh1));
}

// K5: global min/max over elu(g @ W_lin + b_lin)  (for HeteroMinMaxNorm)
__global__ void k_minmax(const float* __restrict__ g, const float* __restrict__ Wl,
                         const float* __restrict__ bl, unsigned* __restrict__ mm,
                         int Mtot) {
  int row = blockIdx.x * blockDim.x + threadIdx.x;
  float mn = 3.4e38f, mx = -3.4e38f;
  if (row < Mtot) {
    float g0 = g[row * 2], g1 = g[row * 2 + 1];
    for (int h = 0; h < 64; ++h) {
      float y = eluf(g0 * Wl[h] + g1 * Wl[64 + h] + bl[h]);
      mn = fminf(mn, y);
      mx = fmaxf(mx, y);
    }
  }
  unsigned emn = f2o(mn), emx = f2o(mx);
  for (int off = 16; off > 0; off >>= 1) {
    unsigned on = __shfl_xor(emn, off, 32);
    unsigned ox = __shfl_xor(emx, off, 32);
    emn = on < emn ? on : emn;
    emx = ox > emx ? ox : emx;
  }
  if ((threadIdx.x & 31) == 0) {
    atomicMin(mm, emn);
    atomicMax(mm + 1, emx);
  }
}

__global__ void k_scal(const unsigned* __restrict__ mm, float* __restrict__ scal) {
  if (threadIdx.x == 0 && blockIdx.x == 0) {
    float mn = o2f(mm[0]), mx = o2f(mm[1]);
    float rmin = 0.1f * mn, rmax = 0.9f + 0.1f * mx;
    float sA = 2.f / (rmax - rmin + 1e-5f);
    scal[0] = sA;
    scal[1] = -rmin * sA - 1.f;
  }
}

// ---------------------------------------------------------------------------
// K6: fused norm -> [M,64]@W_fc[64,64] via V_WMMA_F32_16X16X4_F32 -> elu ->
//     half-sums -> @W_lin2 -> elu -> duplicated output write.
// One wave per 16-row tile; 8 waves/block; W_fc staged in LDS *k-pair
// interleaved* so each B operand is a single aligned 8-byte LDS load.
// ---------------------------------------------------------------------------
__global__ __launch_bounds__(256) void k_fc(const float* __restrict__ g, int gBase,
                                            int M, const float* __restrict__ Wl,
                                            const float* __restrict__ bl,
                                            const float* __restrict__ scal,
                                            const float* __restrict__ Wfc,
                                            const float* __restrict__ bfc,
                                            const float* __restrict__ Wl2,
                                            const float* __restrict__ bl2,
                                            float* __restrict__ out, int outBase) {
  // sW pair-interleaved: pair p = k>>1, sW[(p*64 + col)*2 + (k&1)] = Wfc[k][col]
  __shared__ __align__(16) float sW[4096];
  __shared__ __align__(16) float sT[8][1024];  // per-wave 16x64 tile (A, then fc)
  __shared__ float sS[8][32];                  // per-wave row half sums
  int tid = threadIdx.x;
  int wave = tid >> 5, lane = tid & 31;
  int l16 = lane & 15, half = lane >> 4;
  for (int i = tid; i < 4096; i += 256) {
    int k = i >> 6, col = i & 63;
    sW[(((k >> 1) * 64) + col) * 2 + (k & 1)] = Wfc[i];
  }
  float sA = scal[0], sB = scal[1];
  int row0 = (blockIdx.x * 8 + wave) * 16;
  // Build scaled A tile: A[r][h] = norm(elu(g0*Wl0[h]+g1*Wl1[h]+bl[h]))
  for (int i = lane; i < 1024; i += 32) {
    int r = i >> 6, h = i & 63;
    int rr = row0 + r;
    if (rr >= M) rr = M - 1;  // clamp: keeps control flow uniform
    const float* gp = g + (size_t)(gBase + rr) * 2;
    float y = eluf(gp[0] * Wl[h] + gp[1] * Wl[64 + h] + bl[h]);
    sT[wave][i] = y * sA + sB;
  }
  __syncthreads();
  v8f acc0 = {}, acc1 = {}, acc2 = {}, acc3 = {};
  const float* At = sT[wave];
  const v2f* sWp = (const v2f*)sW;
#pragma unroll
  for (int kk = 0; kk < 16; ++kk) {
    int k0 = kk * 4 + 2 * half;  // lanes 0-15: K=4kk,4kk+1; lanes 16-31: +2,+3
    int kp = 2 * kk + half;      // interleaved pair row
    v2f a = *(const v2f*)(At + l16 * 64 + k0);
    v2f b0 = sWp[kp * 64 + l16];
    v2f b1 = sWp[kp * 64 + 16 + l16];
    v2f b2 = sWp[kp * 64 + 32 + l16];
    v2f b3 = sWp[kp * 64 + 48 + l16];
    acc0 = __builtin_amdgcn_wmma_f32_16x16x4_f32(false, a, false, b0, (short)0, acc0, false, false);
    acc1 = __builtin_amdgcn_wmma_f32_16x16x4_f32(false, a, false, b1, (short)0, acc1, false, false);
    acc2 = __builtin_amdgcn_wmma_f32_16x16x4_f32(false, a, false, b2, (short)0, acc2, false, false);
    acc3 = __builtin_amdgcn_wmma_f32_16x16x4_f32(false, a, false, b3, (short)0, acc3, false, false);
  }
  __syncthreads();
  // D layout: VGPR r -> row (r + 8*half), col = ntile*16 + l16
#pragma unroll
  for (int r = 0; r < 8; ++r) {
    int row = r + 8 * half;
    sT[wave][row * 64 + l16]      = eluf(acc0[r] + bfc[l16]);
    sT[wave][row * 64 + 16 + l16] = eluf(acc1[r] + bfc[16 + l16]);
    sT[wave][row * 64 + 32 + l16] = eluf(acc2[r] + bfc[32 + l16]);
    sT[wave][row * 64 + 48 + l16] = eluf(acc3[r] + bfc[48 + l16]);
  }
  __syncthreads();
  if (lane < 16) {
    float s0 = 0.f, s1 = 0.f;
    for (int h = 0; h < 32; ++h) {
      s0 += sT[wave][l16 * 64 + h];
      s1 += sT[wave][l16 * 64 + 32 + h];
    }
    sS[wave][l16] = s0;
    sS[wave][16 + l16] = s1;
  }
  __syncthreads();
  for (int i = lane; i < 1024; i += 32) {
    int r = i >> 6, h = i & 63;
    int row = row0 + r;
    if (row < M) {
      float v = eluf(sS[wave][r] * Wl2[h] + sS[wave][16 + r] * Wl2[64 + h] + bl2[h]);
      size_t o0 = ((size_t)(outBase + row)) * 64 + h;
      out[o0] = v;
      out[o0 + (size_t)M * 64] = v;  // node_lists has two identical entries
    }
  }
}

// ---------------------------------------------------------------------------
extern "C" void kernel_launch(void* const* d_in, const int* in_sizes, int n_in,
                              void* d_out, int out_size, void* d_ws, size_t ws_size,
                              hipStream_t stream) {
  (void)in_sizes; (void)n_in; (void)out_size; (void)ws_size;
  static const int NS[4] = {4000, 30000, 15000, 12000};          // SB PQ PV NB
  static const int ESrc[15] = {2, 0, 0, 2, 3, 1, 0, 1, 3, 1, 2, 3, 2, 1, 3};
  static const int EDst[15] = {0, 1, 3, 1, 1, 3, 2, 0, 0, 2, 3, 2, 2, 1, 3};

  const float* Wq  = (const float*)d_in[10];
  const float* bq  = (const float*)d_in[11];
  const float* Wk  = (const float*)d_in[12];
  const float* bk  = (const float*)d_in[13];
  const float* Wv  = (const float*)d_in[14];
  const float* bv  = (const float*)d_in[15];
  const float* We  = (const float*)d_in[16];
  const float* be  = (const float*)d_in[17];
  const float* Ws  = (const float*)d_in[18];
  const float* bs  = (const float*)d_in[19];
  const float* Wl  = (const float*)d_in[20];
  const float* bl  = (const float*)d_in[21];
  const float* Wfc = (const float*)d_in[22];
  const float* bfc = (const float*)d_in[23];
  const float* Wl2 = (const float*)d_in[24];
  const float* bl2 = (const float*)d_in[25];

  // workspace layout (floats)
  float* w = (float*)d_ws;
  float* derived = w;                               // 30*192
  unsigned* mm = (unsigned*)(w + 5760);             // 2
  float* scal = w + 5762;                           // 2
  float* dstb = w + 5764;                           // 30000*16
  float* srcb = dstb + 480000;                      // 30000*4
  unsigned* mseg = (unsigned*)(srcb + 120000);      // 30000
  float* den = (float*)mseg + 30000;                // 30000
  float* logit = den + 30000;                       // 150000
  float* pe = logit + 150000;                       // 150000
  float* g = pe + 150000;                           // 480000*2

  int rnk[15], cnt[4] = {0, 0, 0, 0};
  for (int t = 0; t < 15; ++t) rnk[t] = cnt[EDst[t]]++;
  int Gb[4], Ob[4];
  int acc = 0, acco = 0;
  for (int j = 0; j < 4; ++j) {
    Gb[j] = acc;
    int Mj = 2 * cnt[j] * NS[j];
    acc += Mj;
    Ob[j] = acco;
    acco += 2 * Mj;
  }
  const int Mtot = acc;  // 480000

  k_derive<<<30, 160, 0, stream>>>(Wq, bq, Wk, bk, Wv, bv, We, be, Ws, bs, derived, mm);

  const int eb = (EDG + 255) / 256;
  for (int i = 0; i < 2; ++i) {
    for (int t = 0; t < 15; ++t) {
      int it = i * 15 + t;
      int sid = ESrc[t], did = EDst[t];
      const float* xs = (const float*)d_in[2 * sid];
      const float* cs = (const float*)d_in[2 * sid + 1];
      const float* xd = (const float*)d_in[2 * did];
      const float* cd = (const float*)d_in[2 * did + 1];
      const int* srcp = (const int*)d_in[8] + (size_t)t * 2 * EDG;
      const int* dstp = srcp + EDG;
      const float* ea = (const float*)d_in[9] + (size_t)t * EDG * 2;
      int Nd = NS[did], Nsn = NS[sid];
      int gBase = Gb[did] + (i * cnt[did] + rnk[t]) * Nd;
      const float* drec = derived + it * D_STRIDE;
      int nmax = Nd > Nsn ? Nd : Nsn;
      k_prep<<<(nmax + 255) / 256, 256, 0, stream>>>(xd, cd, xs, cs, drec, dstb, srcb,
                                                     mseg, den, g, gBase, Nd, Nsn);
      k_edge1<<<eb, 256, 0, stream>>>(srcp, dstp, ea, xs, cs, dstb, srcb, logit, mseg);
      k_edge2<<<eb, 256, 0, stream>>>(dstp, logit, mseg, pe, den);
      k_edge3<<<eb, 256, 0, stream>>>(srcp, dstp, ea, srcb, den, pe, drec, g, gBase);
    }
  }

  k_minmax<<<(Mtot + 255) / 256, 256, 0, stream>>>(g, Wl, bl, mm, Mtot);
  k_scal<<<1, 32, 0, stream>>>(mm, scal);

  for (int j = 0; j < 4; ++j) {
    int M = 2 * cnt[j] * NS[j];
    int tiles = (M + 15) / 16;
    int grid = (tiles + 7) / 8;
    k_fc<<<grid, 256, 0, stream>>>(g, Gb[j], M, Wl, bl, scal, Wfc + (size_t)j * 4096,
                                   bfc + (size_t)j * 64, Wl2, bl2, (float*)d_out, Ob[j]);
  }
}